// LocalFeatureLoss_9758165696614
// MI455X (gfx1250) — compile-verified
//
#include <hip/hip_runtime.h>
#include <hip/hip_bf16.h>
#include <math.h>

typedef __attribute__((ext_vector_type(2))) float v2f;
typedef __attribute__((ext_vector_type(4))) float v4f;
typedef __attribute__((ext_vector_type(8))) float v8f;

#define BB 8
#define NN 4096
#define KNN 10
#define BN (BB * NN)
#define NT (NN / 16)   // 256 column/row tiles

// Branchless sorted insert of (dv, ci) into ascending top-K lists.
// If dv >= bestd[KNN-1] this is a natural no-op (candidate falls off the end),
// so it is safe to run for lanes that have no hit.
__device__ __forceinline__ void insert10(float dv, int ci,
                                         float (&bestd)[KNN], int (&besti)[KNN]) {
#pragma unroll
  for (int j = 0; j < KNN; ++j) {
    const bool c = dv < bestd[j];
    const float dkeep = c ? dv : bestd[j];
    const float dpass = c ? bestd[j] : dv;
    const int   ikeep = c ? ci : besti[j];
    const int   ipass = c ? besti[j] : ci;
    bestd[j] = dkeep; dv = dpass;
    besti[j] = ikeep; ci = ipass;
  }
}

// ---------------------------------------------------------------------------
// Kernel 1: fused Gram-tile (WMMA f32 16x16x4) + per-row top-10 selection.
// One wave (32 threads) per 16-row tile. Ranking key: |x_j|^2 - 2*G[i][j]
// (row-constant |x_i|^2 dropped; per-row ordering unchanged).
// Guards are wave-uniform ballots -> scalar branches, no EXEC divergence.
// ---------------------------------------------------------------------------
__global__ __launch_bounds__(32) void knn_kernel(const float* __restrict__ xyz,
                                                 int* __restrict__ knn_idx) {
  __shared__ float tile[16 * 16];

  const int b    = blockIdx.x >> 8;     // / NT
  const int rt   = blockIdx.x & 255;    // row tile
  const int lane = threadIdx.x;         // 0..31 (wave32)
  const float* __restrict__ P = xyz + (size_t)b * NN * 3;

  // --- A fragment (16x4, K=3 padded to 4) ---
  // lanes 0-15: (x,y) of row M=lane ; lanes 16-31: (z,0) of row M=lane-16
  const int am   = lane & 15;
  const int arow = rt * 16 + am;
  v2f A;
  if (lane < 16) { A.x = P[arow * 3 + 0]; A.y = P[arow * 3 + 1]; }
  else           { A.x = P[arow * 3 + 2]; A.y = 0.0f;            }

  // per-lane top-10 state (sorted ascending; [9] is current worst)
  float bestd[KNN];
  int   besti[KNN];
#pragma unroll
  for (int j = 0; j < KNN; ++j) { bestd[j] = 3.4e38f; besti[j] = 0; }

  const int cn    = lane & 15;          // column within tile handled by this lane
  const int rbase = (lane >> 4) << 3;   // 0 or 8 (row offset of my C/D half)
  const int selr  = lane >> 1;          // selection: 2 lanes per row
  const int selh  = lane & 1;           // which half of the 16 columns
  const float* __restrict__ myrow = tile + selr * 16 + selh * 8; // 32B aligned

  for (int t = 0; t < NT; ++t) {
    const int col = t * 16 + cn;
    const float cx = P[col * 3 + 0];
    const float cy = P[col * 3 + 1];
    const float cz = P[col * 3 + 2];
    const float sqc = cx * cx + cy * cy + cz * cz;

    // B fragment (4x16), symmetric to A layout
    v2f Bf;
    if (lane < 16) { Bf.x = cx; Bf.y = cy;   }
    else           { Bf.x = cz; Bf.y = 0.0f; }

    v8f Cacc = {};
    Cacc = __builtin_amdgcn_wmma_f32_16x16x4_f32(
        /*neg_a=*/false, A, /*neg_b=*/false, Bf,
        /*c_mod=*/(short)0, Cacc, /*reuse_a=*/false, /*reuse_b=*/false);

    // ranking key -> LDS (remap from N-per-lane to row-major tile)
#pragma unroll
    for (int j = 0; j < 8; ++j)
      tile[(rbase + j) * 16 + cn] = sqc - 2.0f * Cacc[j];
    __syncthreads();

    // selection: two b128 LDS loads + fmin-tree pre-screen; wave-uniform
    // ballot guards so skipped paths cost one scalar branch each.
    const v4f q0 = ((const v4f*)myrow)[0];
    const v4f q1 = ((const v4f*)myrow)[1];
    const float mn = fminf(fminf(fminf(q0.x, q0.y), fminf(q0.z, q0.w)),
                           fminf(fminf(q1.x, q1.y), fminf(q1.z, q1.w)));
    if (__builtin_amdgcn_ballot_w32(mn < bestd[KNN - 1])) {
      const int cbase = t * 16 + selh * 8;
      const float cand[8] = {q0.x, q0.y, q0.z, q0.w, q1.x, q1.y, q1.z, q1.w};
#pragma unroll
      for (int i = 0; i < 8; ++i) {
        if (__builtin_amdgcn_ballot_w32(cand[i] < bestd[KNN - 1])) {
          insert10(cand[i], cbase + i, bestd, besti);
        }
      }
    }
    __syncthreads();
  }

  // merge the two half-lists of each row: snapshot partner first, then insert
  float od[KNN]; int oi[KNN];
#pragma unroll
  for (int j = 0; j < KNN; ++j) {
    od[j] = __shfl_xor(bestd[j], 1, 32);
    oi[j] = __shfl_xor(besti[j], 1, 32);
  }
#pragma unroll
  for (int j = 0; j < KNN; ++j) {
    insert10(od[j], oi[j], bestd, besti);
  }

  if (selh == 0) {
    const size_t base = ((size_t)b * NN + rt * 16 + selr) * KNN;
#pragma unroll
    for (int j = 0; j < KNN; ++j) knn_idx[base + j] = besti[j];
  }
}

// ---------------------------------------------------------------------------
// Kernel 2: per-point PCA normal (smallest eigenvector of 3x3 covariance,
// analytic trig method) + signed point-to-plane distance.
// ---------------------------------------------------------------------------
__global__ void ptof_kernel(const float* __restrict__ xyz,
                            const int* __restrict__ knn_idx,
                            float* __restrict__ ptof) {
  const int gid = blockIdx.x * blockDim.x + threadIdx.x;
  if (gid >= BN) return;
  const int b = gid >> 12;      // / NN
  const int n = gid & (NN - 1);
  const float* __restrict__ P = xyz + (size_t)b * NN * 3;

  float px[KNN], py[KNN], pz[KNN];
  float mx = 0.f, my = 0.f, mz = 0.f;
#pragma unroll
  for (int j = 0; j < KNN; ++j) {
    const int idx = knn_idx[(size_t)gid * KNN + j];
    px[j] = P[idx * 3 + 0]; py[j] = P[idx * 3 + 1]; pz[j] = P[idx * 3 + 2];
    mx += px[j]; my += py[j]; mz += pz[j];
  }
  const float inv = 1.0f / (float)KNN;
  mx *= inv; my *= inv; mz *= inv;

  float cxx = 0.f, cxy = 0.f, cxz = 0.f, cyy = 0.f, cyz = 0.f, czz = 0.f;
#pragma unroll
  for (int j = 0; j < KNN; ++j) {
    const float dx = px[j] - mx, dy = py[j] - my, dz = pz[j] - mz;
    cxx += dx * dx; cxy += dx * dy; cxz += dx * dz;
    cyy += dy * dy; cyz += dy * dz; czz += dz * dz;
  }

  // smallest eigenvalue of symmetric 3x3 (trigonometric method)
  const float p1 = cxy * cxy + cxz * cxz + cyz * cyz;
  const float q  = (cxx + cyy + czz) * (1.0f / 3.0f);
  const float p2 = (cxx - q) * (cxx - q) + (cyy - q) * (cyy - q) +
                   (czz - q) * (czz - q) + 2.0f * p1;
  const float p  = sqrtf(p2 * (1.0f / 6.0f));
  float lam = q;
  if (p > 1e-20f) {
    const float ip  = 1.0f / p;
    const float b00 = (cxx - q) * ip, b11 = (cyy - q) * ip, b22 = (czz - q) * ip;
    const float b01 = cxy * ip, b02 = cxz * ip, b12 = cyz * ip;
    float detB = b00 * (b11 * b22 - b12 * b12)
               - b01 * (b01 * b22 - b12 * b02)
               + b02 * (b01 * b12 - b11 * b02);
    float r = fminf(1.0f, fmaxf(-1.0f, detB * 0.5f));
    const float phi = acosf(r) * (1.0f / 3.0f);
    lam = q + 2.0f * p * cosf(phi + 2.0943951023931953f); // +2*pi/3 -> smallest
  }

  // eigenvector: best cross product of rows of (C - lam*I)
  const float a00 = cxx - lam, a11 = cyy - lam, a22 = czz - lam;
  float v0x = cxy * cyz - cxz * a11, v0y = cxz * cxy - a00 * cyz, v0z = a00 * a11 - cxy * cxy;
  float v1x = cxy * a22 - cxz * cyz, v1y = cxz * cxz - a00 * a22, v1z = a00 * cyz - cxy * cxz;
  float v2x = a11 * a22 - cyz * cyz, v2y = cyz * cxz - cxy * a22, v2z = cxy * cyz - a11 * cxz;
  const float n0 = v0x * v0x + v0y * v0y + v0z * v0z;
  const float n1 = v1x * v1x + v1y * v1y + v1z * v1z;
  const float n2 = v2x * v2x + v2y * v2y + v2z * v2z;
  float nx, ny, nz, nrm;
  if (n0 >= n1 && n0 >= n2)      { nx = v0x; ny = v0y; nz = v0z; nrm = n0; }
  else if (n1 >= n2)             { nx = v1x; ny = v1y; nz = v1z; nrm = n1; }
  else                           { nx = v2x; ny = v2y; nz = v2z; nrm = n2; }
  if (nrm > 1e-30f) {
    const float rn = rsqrtf(nrm);
    nx *= rn; ny *= rn; nz *= rn;
  } else { nx = 0.f; ny = 0.f; nz = 1.f; }

  const float qx = P[n * 3 + 0] - mx;
  const float qy = P[n * 3 + 1] - my;
  const float qz = P[n * 3 + 2] - mz;
  ptof[gid] = qx * nx + qy * ny + qz * nz;
}

// ---------------------------------------------------------------------------
// Kernel 3: deterministic single-block reduction -> scalar loss.
// ---------------------------------------------------------------------------
__global__ __launch_bounds__(256) void loss_kernel(const float* __restrict__ p1,
                                                   const float* __restrict__ p2,
                                                   float* __restrict__ out) {
  __shared__ float s1[256];
  __shared__ float s2[256];
  float a1 = 0.f, a2 = 0.f;
  for (int i = threadIdx.x; i < BN; i += 256) {
    const float x = p1[i], y = p2[i];
    const float d = fabsf(x) - fabsf(y);
    a1 += d * d;
    const float bent = fmaxf(y - x, 0.0f);
    a2 += bent * bent;
  }
  s1[threadIdx.x] = a1;
  s2[threadIdx.x] = a2;
  __syncthreads();
  for (int s = 128; s > 0; s >>= 1) {
    if (threadIdx.x < s) {
      s1[threadIdx.x] += s1[threadIdx.x + s];
      s2[threadIdx.x] += s2[threadIdx.x + s];
    }
    __syncthreads();
  }
  if (threadIdx.x == 0)
    out[0] = (s1[0] + 5.0f * s2[0]) * (1.0f / (float)BN);
}

extern "C" void kernel_launch(void* const* d_in, const int* in_sizes, int n_in,
                              void* d_out, int out_size, void* d_ws, size_t ws_size,
                              hipStream_t stream) {
  const float* xyz1 = (const float*)d_in[0];
  const float* xyz2 = (const float*)d_in[1];

  char* ws = (char*)d_ws;
  int*   knn   = (int*)ws;                                   // BN*KNN ints
  float* ptof1 = (float*)(ws + (size_t)BN * KNN * sizeof(int));
  float* ptof2 = ptof1 + BN;

  knn_kernel<<<BB * NT, 32, 0, stream>>>(xyz1, knn);
  ptof_kernel<<<(BN + 255) / 256, 256, 0, stream>>>(xyz1, knn, ptof1);
  ptof_kernel<<<(BN + 255) / 256, 256, 0, stream>>>(xyz2, knn, ptof2);
  loss_kernel<<<1, 256, 0, stream>>>(ptof1, ptof2, (float*)d_out);
}